// GAT_1881195675685
// MI455X (gfx1250) — compile-verified
//
#include <hip/hip_runtime.h>
#include <hip/hip_bf16.h>

#define N_NODES 50000
#define N_EDGES 800000
#define EP (N_EDGES + N_NODES)
#define NEG_SLOPE 0.2f
#define LN_EPS 1e-5f
#define LDST 260   // 256 + 4 pad: keeps 16B alignment, spreads LDS banks

typedef __attribute__((ext_vector_type(2))) float v2f;
typedef __attribute__((ext_vector_type(8))) float v8f;

// ---------------------------------------------------------------------------
// GEMM  C[M x NC] = A[M x 256] * B[256 x NC]   via V_WMMA_F32_16X16X4_F32.
// block = 256 threads (8 waves); each wave owns one 16x16 output tile,
// waves tile 128 columns; blockIdx.x tiles rows (M divisible by 16),
// blockIdx.y tiles column groups of 128.
// A-tile (16x256 f32) is staged to LDS with GLOBAL_LOAD_ASYNC_TO_LDS_B128
// (CDNA5 async memory->LDS path, ASYNCcnt-tracked, no VGPR round-trip).
// ---------------------------------------------------------------------------
template <int NC>
__global__ __launch_bounds__(256) void gat_gemm_f32wmma(
    const float* __restrict__ A, const float* __restrict__ B,
    float* __restrict__ C)
{
  __shared__ float As[16 * LDST];
  const int tid  = threadIdx.x;
  const int row0 = blockIdx.x * 16;

  // async A-tile stage: 16 rows x 256 floats; 64 bytes per thread
  {
    const int r  = tid >> 4;          // 0..15  (row of tile)
    const int c0 = (tid & 15) * 16;   // 0..240 (first float of chunk)
    const unsigned long long ga =
        (unsigned long long)(const void*)(A + (size_t)(row0 + r) * 256 + c0);
    const unsigned int la =
        (unsigned int)(unsigned long long)(const void*)(As + r * LDST + c0);
    // IOFFSET is added to both the LDS and global addresses (ISA 10.7/15.18)
    asm volatile(
        "global_load_async_to_lds_b128 %0, %1, off\n\t"
        "global_load_async_to_lds_b128 %0, %1, off offset:16\n\t"
        "global_load_async_to_lds_b128 %0, %1, off offset:32\n\t"
        "global_load_async_to_lds_b128 %0, %1, off offset:48"
        :: "v"(la), "v"(ga) : "memory");
    asm volatile("s_wait_asynccnt 0x0" ::: "memory");
  }
  __syncthreads();

  const int wave = tid >> 5;
  const int lane = tid & 31;
  const int ln   = lane & 15;   // row (A) / col (B) within 16x16 tile
  const int kh   = lane >> 4;   // k-half select
  const int col  = blockIdx.y * 128 + wave * 16 + ln;
  const float* Bp = B + col;

  v8f acc = {};
#pragma unroll 4
  for (int kk = 0; kk < 64; ++kk) {
    const int k = kk * 4 + kh * 2;
    v2f a = *reinterpret_cast<const v2f*>(As + ln * LDST + k);   // ds_load_b64
    v2f b;
    b.x = Bp[k * NC];
    b.y = Bp[(k + 1) * NC];
    acc = __builtin_amdgcn_wmma_f32_16x16x4_f32(
        false, a, false, b, (short)0, acc, false, false);
  }
#pragma unroll
  for (int j = 0; j < 8; ++j)
    C[(size_t)(row0 + j + 8 * kh) * NC + col] = acc[j];
}

// ---------------------------------------------------------------------------
// attention dot products, layer 1: 8 heads x 32 ch. block = node, wave = head
// ---------------------------------------------------------------------------
__global__ __launch_bounds__(256) void att_heads8(
    const float* __restrict__ h, const float* __restrict__ ws,
    const float* __restrict__ wd, float* __restrict__ asrc,
    float* __restrict__ adst)
{
  const int n    = blockIdx.x;
  const int hd   = threadIdx.x >> 5;
  const int lane = threadIdx.x & 31;
  const float hv = h[(size_t)n * 256 + hd * 32 + lane];
  float s = hv * ws[hd * 32 + lane];
  float d = hv * wd[hd * 32 + lane];
  for (int m = 16; m; m >>= 1) { s += __shfl_xor(s, m); d += __shfl_xor(d, m); }
  if (lane == 0) { asrc[n * 8 + hd] = s; adst[n * 8 + hd] = d; }
}

// attention dot products, layer 2: 1 head x 128 ch. wave = node
__global__ __launch_bounds__(256) void att_head1(
    const float* __restrict__ h, const float* __restrict__ ws,
    const float* __restrict__ wd, float* __restrict__ asrc,
    float* __restrict__ adst)
{
  const int gw   = (blockIdx.x * 256 + threadIdx.x) >> 5;
  const int lane = threadIdx.x & 31;
  if (gw >= N_NODES) return;
  float s = 0.f, d = 0.f;
#pragma unroll
  for (int i = 0; i < 4; ++i) {
    const float v = h[(size_t)gw * 128 + i * 32 + lane];
    s += v * ws[i * 32 + lane];
    d += v * wd[i * 32 + lane];
  }
  for (int m = 16; m; m >>= 1) { s += __shfl_xor(s, m); d += __shfl_xor(d, m); }
  if (lane == 0) { asrc[gw] = s; adst[gw] = d; }
}

// ---------------------------------------------------------------------------
// edge pass A: ex = exp(leaky_relu(a_src[s]+a_dst[d])); denom[d] += ex
// (max-shift omitted: logits are O(0.1); softmax ratio unchanged)
// ---------------------------------------------------------------------------
__global__ __launch_bounds__(256) void edge_softmax8(
    const int* __restrict__ ei, const float* __restrict__ asrc,
    const float* __restrict__ adst, float* __restrict__ ex,
    float* __restrict__ den)
{
  const int e = blockIdx.x * 256 + threadIdx.x;
  if (e >= EP) return;
  int s, d;
  if (e < N_EDGES) { s = ei[e]; d = ei[N_EDGES + e]; } else { s = d = e - N_EDGES; }
#pragma unroll
  for (int hd = 0; hd < 8; ++hd) {
    float v = asrc[s * 8 + hd] + adst[d * 8 + hd];
    v = v > 0.f ? v : NEG_SLOPE * v;
    const float w = __expf(v);
    ex[(size_t)e * 8 + hd] = w;
    atomicAdd(&den[d * 8 + hd], w);
  }
}

__global__ __launch_bounds__(256) void edge_softmax1h(
    const int* __restrict__ ei, const float* __restrict__ asrc,
    const float* __restrict__ adst, float* __restrict__ ex,
    float* __restrict__ den)
{
  const int e = blockIdx.x * 256 + threadIdx.x;
  if (e >= EP) return;
  int s, d;
  if (e < N_EDGES) { s = ei[e]; d = ei[N_EDGES + e]; } else { s = d = e - N_EDGES; }
  float v = asrc[s] + adst[d];
  v = v > 0.f ? v : NEG_SLOPE * v;
  const float w = __expf(v);
  ex[e] = w;
  atomicAdd(&den[d], w);
}

// ---------------------------------------------------------------------------
// edge pass B: out[d, :] += h[s, :] * (ex[e]/denom[d]).  1 block = 1 edge,
// thread = channel -> fully coalesced gather + coalesced L2 float atomics.
// ---------------------------------------------------------------------------
__global__ __launch_bounds__(256) void edge_aggr8(
    const int* __restrict__ ei, const float* __restrict__ hf,
    const float* __restrict__ ex, const float* __restrict__ den,
    float* __restrict__ outp)
{
  const int e = blockIdx.x;
  const int t = threadIdx.x;
  int s, d;
  if (e < N_EDGES) { s = ei[e]; d = ei[N_EDGES + e]; } else { s = d = e - N_EDGES; }
  const int hd = t >> 5;
  const float alpha = ex[(size_t)e * 8 + hd] / (den[d * 8 + hd] + 1e-16f);
  atomicAdd(&outp[(size_t)d * 256 + t], hf[(size_t)s * 256 + t] * alpha);
}

__global__ __launch_bounds__(128) void edge_aggr1h(
    const int* __restrict__ ei, const float* __restrict__ hf,
    const float* __restrict__ ex, const float* __restrict__ den,
    float* __restrict__ outp)
{
  const int e = blockIdx.x;
  const int t = threadIdx.x;
  int s, d;
  if (e < N_EDGES) { s = ei[e]; d = ei[N_EDGES + e]; } else { s = d = e - N_EDGES; }
  const float alpha = ex[e] / (den[d] + 1e-16f);
  atomicAdd(&outp[(size_t)d * 128 + t], hf[(size_t)s * 128 + t] * alpha);
}

__global__ __launch_bounds__(256) void bias_relu(
    float* __restrict__ x, const float* __restrict__ b, int total, int mask)
{
  const int i = blockIdx.x * 256 + threadIdx.x;
  if (i < total) {
    const float v = x[i] + b[i & mask];
    x[i] = v > 0.f ? v : 0.f;
  }
}

// ---------------------------------------------------------------------------
// BatchNorm column stats: thread = feature, blocks stride over rows
// ---------------------------------------------------------------------------
__global__ __launch_bounds__(128) void bn_stats(
    const float* __restrict__ x, float* __restrict__ sum, float* __restrict__ sq)
{
  const int t = threadIdx.x;
  float s = 0.f, q = 0.f;
  for (int r = blockIdx.x; r < N_NODES; r += gridDim.x) {
    const float v = x[(size_t)r * 128 + t];
    s += v; q += v * v;
  }
  atomicAdd(&sum[t], s);
  atomicAdd(&sq[t], q);
}

// fused BatchNorm (precomputed stats) + LayerNorm; block = node
__global__ __launch_bounds__(128) void bn_ln(
    const float* __restrict__ x, const float* __restrict__ sum,
    const float* __restrict__ sq, const float* __restrict__ bg,
    const float* __restrict__ bb, const float* __restrict__ lg,
    const float* __restrict__ lb, float* __restrict__ outp)
{
  __shared__ float red[8];
  const int n = blockIdx.x, t = threadIdx.x;
  const float inv_n = 1.0f / (float)N_NODES;
  const float mu  = sum[t] * inv_n;
  const float var = sq[t] * inv_n - mu * mu;
  float v = (x[(size_t)n * 128 + t] - mu) * rsqrtf(var + LN_EPS) * bg[t] + bb[t];

  float s = v, q = v * v;
  for (int m = 16; m; m >>= 1) { s += __shfl_xor(s, m); q += __shfl_xor(q, m); }
  const int w = t >> 5, lane = t & 31;
  if (lane == 0) { red[w] = s; red[4 + w] = q; }
  __syncthreads();
  s = red[0] + red[1] + red[2] + red[3];
  q = red[4] + red[5] + red[6] + red[7];
  const float mu2  = s * (1.0f / 128.0f);
  const float var2 = q * (1.0f / 128.0f) - mu2 * mu2;
  outp[(size_t)n * 128 + t] = (v - mu2) * rsqrtf(var2 + LN_EPS) * lg[t] + lb[t];
}

// ---------------------------------------------------------------------------
extern "C" void kernel_launch(void* const* d_in, const int* in_sizes, int n_in,
                              void* d_out, int out_size, void* d_ws, size_t ws_size,
                              hipStream_t stream)
{
  (void)in_sizes; (void)n_in; (void)out_size; (void)ws_size;
  const float* x     = (const float*)d_in[0];
  const int*   ei    = (const int*)d_in[1];
  const float* W1    = (const float*)d_in[2];
  const float* atts1 = (const float*)d_in[3];
  const float* attd1 = (const float*)d_in[4];
  const float* b1    = (const float*)d_in[5];
  const float* W2    = (const float*)d_in[6];
  const float* atts2 = (const float*)d_in[7];
  const float* attd2 = (const float*)d_in[8];
  const float* b2    = (const float*)d_in[9];
  const float* bng   = (const float*)d_in[10];
  const float* bnb   = (const float*)d_in[11];
  const float* lng   = (const float*)d_in[12];
  const float* lnb   = (const float*)d_in[13];
  float* out = (float*)d_out;

  // workspace layout (aliased across the two layers)
  float* h1    = (float*)d_ws;                       // N*256
  float* out1  = h1   + (size_t)N_NODES * 256;       // N*256
  float* ex1   = out1 + (size_t)N_NODES * 256;       // EP*8
  float* den1  = ex1  + (size_t)EP * 8;              // N*8
  float* asr   = den1 + (size_t)N_NODES * 8;         // N*8
  float* ads   = asr  + (size_t)N_NODES * 8;         // N*8
  float* bnsum = ads  + (size_t)N_NODES * 8;         // 128
  float* bnsq  = bnsum + 128;                        // 128
  float* h2 = h1; float* out2 = out1; float* ex2 = ex1; float* den2 = den1;

  const dim3 blk(256);

  // ---- layer 1 (8 heads x 32 ch, concat) ----
  gat_gemm_f32wmma<256><<<dim3(N_NODES / 16, 2), blk, 0, stream>>>(x, W1, h1);
  att_heads8<<<dim3(N_NODES), blk, 0, stream>>>(h1, atts1, attd1, asr, ads);
  hipMemsetAsync(den1, 0, (size_t)N_NODES * 8 * sizeof(float), stream);
  edge_softmax8<<<dim3((EP + 255) / 256), blk, 0, stream>>>(ei, asr, ads, ex1, den1);
  hipMemsetAsync(out1, 0, (size_t)N_NODES * 256 * sizeof(float), stream);
  edge_aggr8<<<dim3(EP), blk, 0, stream>>>(ei, h1, ex1, den1, out1);
  bias_relu<<<dim3((N_NODES * 256) / 256), blk, 0, stream>>>(out1, b1, N_NODES * 256, 255);

  // ---- layer 2 (1 head x 128 ch, mean == identity) ----
  gat_gemm_f32wmma<128><<<dim3(N_NODES / 16, 1), blk, 0, stream>>>(out1, W2, h2);
  att_head1<<<dim3((N_NODES + 7) / 8), blk, 0, stream>>>(h2, atts2, attd2, asr, ads);
  hipMemsetAsync(den2, 0, (size_t)N_NODES * sizeof(float), stream);
  edge_softmax1h<<<dim3((EP + 255) / 256), blk, 0, stream>>>(ei, asr, ads, ex2, den2);
  hipMemsetAsync(out2, 0, (size_t)N_NODES * 128 * sizeof(float), stream);
  edge_aggr1h<<<dim3(EP), dim3(128), 0, stream>>>(ei, h2, ex2, den2, out2);
  bias_relu<<<dim3((N_NODES * 128) / 256), blk, 0, stream>>>(out2, b2, N_NODES * 128, 127);

  // ---- BatchNorm + LayerNorm ----
  hipMemsetAsync(bnsum, 0, 256 * sizeof(float), stream);
  bn_stats<<<dim3(256), dim3(128), 0, stream>>>(out2, bnsum, bnsq);
  bn_ln<<<dim3(N_NODES), dim3(128), 0, stream>>>(out2, bnsum, bnsq, bng, bnb, lng, lnb, out);
}